// DynamicRoIAlign_26070451486998
// MI455X (gfx1250) — compile-verified
//
#include <hip/hip_runtime.h>
#include <hip/hip_bf16.h>
#include <stdint.h>

typedef unsigned int u32;
typedef uint64_t u64;
typedef u32 v4u __attribute__((ext_vector_type(4)));
typedef int  v8i __attribute__((ext_vector_type(8)));
typedef int  v4i __attribute__((ext_vector_type(4)));

#define HF 200
#define WF 200
#define CCH 256
#define OH 14
#define OW 14
#define NPIX (OH * OW)            // 196
#define CSPLIT 4
#define CPB (CCH / CSPLIT)        // 64 channels per block
#define TILE_MAX 104              // max ROI rectangle side (+slack), elements
#define BUF_ELEMS (TILE_MAX * TILE_MAX)   // 10816 floats per buffer
#define NTHREADS 224              // 7 wave32s; 196 active pixel lanes

__device__ __forceinline__ int imin_(int a, int b) { return a < b ? a : b; }
__device__ __forceinline__ int imax_(int a, int b) { return a > b ? a : b; }

// Issue one TDM 2D-tile load: global (w x h rect, row stride WF floats) -> LDS.
// D# packing per CDNA5 ISA 8.3/8.4 (group0: count|lds_addr|global_addr|type=2,
// group1: data_size=4B, tensor_dim0/1 = tile_dim0/1 = w/h, stride0 = WF).
// This toolchain's builtin takes 6 args: (v4u, v8i, v4i, v4i, v8i, i32 cpol);
// the trailing v8i operand is zero-filled (probe-verified form).
__device__ __forceinline__ void tdm_load_tile(u32 lds_byte_off, u64 gaddr, int w, int h) {
  v4u g0;
  g0[0] = 1u;                                             // count=1, user mode
  g0[1] = lds_byte_off;                                   // lds_addr
  g0[2] = (u32)(gaddr & 0xFFFFFFFFull);                   // global_addr[31:0]
  g0[3] = ((u32)((gaddr >> 32) & 0x01FFFFFFull)) | (2u << 30); // addr[56:32] | type=2
  v8i g1;
  g1[0] = (int)(2u << 16);                                // wg_mask=0, data_size=2 (4B)
  g1[1] = (int)((w & 0xFFFF) << 16);                      // tensor_dim0 lo16
  g1[2] = (int)(((u32)(w >> 16) & 0xFFFFu) | ((u32)(h & 0xFFFF) << 16)); // td0 hi | td1 lo
  g1[3] = (int)(((u32)(h >> 16) & 0xFFFFu) | ((u32)(w & 0xFFFF) << 16)); // td1 hi | tile_dim0
  g1[4] = (int)(h & 0xFFFF);                              // tile_dim1 = h, tile_dim2 = 0 (2D)
  g1[5] = WF;                                             // tensor_dim0_stride = 200
  g1[6] = 0;
  g1[7] = 0;
  v4i g2 = {0, 0, 0, 0};
  v4i g3 = {0, 0, 0, 0};
  v8i g4 = {0, 0, 0, 0, 0, 0, 0, 0};
  __builtin_amdgcn_tensor_load_to_lds(g0, g1, g2, g3, g4, 0);
}

extern "C" __global__ void __launch_bounds__(NTHREADS)
roi_align_tdm_kernel(const float* __restrict__ feat,
                     const float* __restrict__ rois,
                     float* __restrict__ out, int K)
{
  extern __shared__ float lds[];   // 2 * BUF_ELEMS floats (double buffer)

  const int k    = blockIdx.x;
  const int cseg = blockIdx.y;
  if (k >= K) return;
  const int t = (int)threadIdx.x;

  // ---- ROI parameters (uniform across block) ----
  const float rb  = rois[k * 5 + 0];
  const float x1  = rois[k * 5 + 1] * 200.0f;
  const float y1  = rois[k * 5 + 2] * 200.0f;
  const float x2  = rois[k * 5 + 3] * 200.0f;
  const float y2  = rois[k * 5 + 4] * 200.0f;
  const float rw  = x2 - x1;
  const float rh  = y2 - y1;
  const int   b   = (int)rb;

  // Replicate reference f32 math exactly (ALIGNED=False path):
  //   fx = x1 + (px/13)*rw ; gx = fx/W*2-1 ; ix = ((gx+1)*W - 1)*0.5
  auto map_x = [&](float px) -> float {
    float fx = x1 + (px * (1.0f / 13.0f)) * rw;
    float gx = fx / 200.0f * 2.0f - 1.0f;
    return ((gx + 1.0f) * 200.0f - 1.0f) * 0.5f;
  };
  auto map_y = [&](float py) -> float {
    float fy = y1 + (py * (1.0f / 13.0f)) * rh;
    float gy = fy / 200.0f * 2.0f - 1.0f;
    return ((gy + 1.0f) * 200.0f - 1.0f) * 0.5f;
  };

  // ---- ROI bounding rectangle in the feature plane (monotone affine map) ----
  const float ixa = map_x(0.0f),  ixb = map_x(13.0f);
  const float iya = map_y(0.0f),  iyb = map_y(13.0f);
  int xlo = imin_(imax_((int)floorf(ixa), 0), WF - 1);
  int ylo = imin_(imax_((int)floorf(iya), 0), HF - 1);
  int xhi = imin_((int)floorf(ixb) + 1, WF - 1);
  int yhi = imin_((int)floorf(iyb) + 1, HF - 1);
  int w = imin_(imax_(xhi - xlo + 1, 1), TILE_MAX);
  int h = imin_(imax_(yhi - ylo + 1, 1), TILE_MAX);
  // Force wave-uniform scalars for the TDM descriptor.
  xlo = __builtin_amdgcn_readfirstlane(xlo);
  ylo = __builtin_amdgcn_readfirstlane(ylo);
  w   = __builtin_amdgcn_readfirstlane(w);
  h   = __builtin_amdgcn_readfirstlane(h);
  const int bu = __builtin_amdgcn_readfirstlane(b);

  // ---- Per-pixel geometry, kept in registers across the channel loop ----
  const int px = t % OW;
  const int py = t / OW;
  const float ix = map_x((float)px);
  const float iy = map_y((float)py);
  const float xf = floorf(ix), yf = floorf(iy);
  const float wx1 = ix - xf, wx0 = 1.0f - wx1;
  const float wy1 = iy - yf, wy0 = 1.0f - wy1;
  const int x0 = (int)xf, y0 = (int)yf;
  const float ax0 = (x0     >= 0 && x0     < WF) ? wx0 : 0.0f;
  const float ax1 = (x0 + 1 >= 0 && x0 + 1 < WF) ? wx1 : 0.0f;
  const float ay0 = (y0     >= 0 && y0     < HF) ? wy0 : 0.0f;
  const float ay1 = (y0 + 1 >= 0 && y0 + 1 < HF) ? wy1 : 0.0f;
  const int lx0 = imin_(imax_(x0     - xlo, 0), w - 1);
  const int lx1 = imin_(imax_(x0 + 1 - xlo, 0), w - 1);
  const int ly0 = imin_(imax_(y0     - ylo, 0), h - 1);
  const int ly1 = imin_(imax_(y0 + 1 - ylo, 0), h - 1);
  const int i00 = ly0 * w + lx0, i01 = ly0 * w + lx1;
  const int i10 = ly1 * w + lx0, i11 = ly1 * w + lx1;

  // ---- TDM double-buffered channel pipeline ----
  const int  c0      = cseg * CPB;
  const u64  fbase   = (u64)(uintptr_t)feat;
  const u32  ldsbase = (u32)(uintptr_t)(&lds[0]);      // LDS byte offset (addr[31:0])
  const u64  gstep   = 4ull * (u64)HF * (u64)WF;       // channel stride in bytes
  u64 ga0 = fbase + 4ull * (((((u64)bu * CCH + (u64)c0) * HF) + (u64)ylo) * WF + (u64)xlo);

  const bool issuer = (t < 32);                        // wave 0 only (wave-uniform branch)
  if (issuer) {
    tdm_load_tile(ldsbase,                          ga0,         w, h);
    tdm_load_tile(ldsbase + (u32)(BUF_ELEMS * 4),   ga0 + gstep, w, h);
  }

  const size_t obase = ((size_t)k * CCH + (size_t)c0) * NPIX + (size_t)t;

  for (int s = 0; s < CPB; ++s) {
    if (issuer) {
      if (s + 1 < CPB) __builtin_amdgcn_s_wait_tensorcnt(1);  // stage s landed
      else             __builtin_amdgcn_s_wait_tensorcnt(0);
    }
    __syncthreads();                                  // tile s visible to all waves

    if (t < NPIX) {
      const float* buf = &lds[(size_t)(s & 1) * BUF_ELEMS];
      const float f00 = buf[i00];
      const float f01 = buf[i01];
      const float f10 = buf[i10];
      const float f11 = buf[i11];
      const float v = ay0 * (ax0 * f00 + ax1 * f01) + ay1 * (ax0 * f10 + ax1 * f11);
      // Output is streamed once (200 MB): NT store keeps features hot in L2.
      __builtin_nontemporal_store(v, &out[obase + (size_t)s * NPIX]);
    }
    __syncthreads();                                  // all reads of buf[s&1] done

    if (issuer && (s + 2 < CPB)) {
      tdm_load_tile(ldsbase + (u32)((s & 1) * BUF_ELEMS * 4),
                    ga0 + (u64)(s + 2) * gstep, w, h);
    }
  }
}

extern "C" void kernel_launch(void* const* d_in, const int* in_sizes, int n_in,
                              void* d_out, int out_size, void* d_ws, size_t ws_size,
                              hipStream_t stream) {
  (void)n_in; (void)out_size; (void)d_ws; (void)ws_size;
  const float* feat = (const float*)d_in[0];
  const float* rois = (const float*)d_in[1];
  float* out = (float*)d_out;
  const int K = in_sizes[1] / 5;

  const size_t shmem = (size_t)2 * BUF_ELEMS * sizeof(float);  // ~86.5 KB
  hipError_t err = hipFuncSetAttribute((const void*)roi_align_tdm_kernel,
                                       hipFuncAttributeMaxDynamicSharedMemorySize,
                                       (int)shmem);
  (void)err;

  dim3 grid((unsigned)K, CSPLIT, 1);
  roi_align_tdm_kernel<<<grid, NTHREADS, shmem, stream>>>(feat, rois, out, K);
}